// GCNModel_RES_46986942218444
// MI455X (gfx1250) — compile-verified
//
#include <hip/hip_runtime.h>

typedef __attribute__((ext_vector_type(2))) float v2f;
typedef __attribute__((ext_vector_type(8))) float v8f;

#define N_NODES 50000
#define D_IN    128
#define D_HID   64
#define D_OUT   128

// ---------------------------------------------------------------- degree ----
__global__ void k_init_deg(float* __restrict__ deg, int n) {
    int i = blockIdx.x * blockDim.x + threadIdx.x;
    if (i < n) deg[i] = 1.0f;  // self-loop contribution
}

__global__ void k_deg(const long long* __restrict__ dst, float* __restrict__ deg, int E) {
    int e = blockIdx.x * blockDim.x + threadIdx.x;
    if (e < E) atomicAdd(&deg[(int)dst[e]], 1.0f);
}

__global__ void k_dinv(const float* __restrict__ deg, float* __restrict__ dinv, int n) {
    int i = blockIdx.x * blockDim.x + threadIdx.x;
    if (i < n) dinv[i] = rsqrtf(fmaxf(deg[i], 1.0f));
}

// ------------------------------------------------------- GEMM1: h0 = x@W1 ---
// One 16x16 output tile per wave32. K=128 -> 32 chained f32 WMMA ops.
// Epilogue: hs = h0*dinv[row]; acc1 = h0*dinv[row]^2 (self-loop seed).
__global__ void k_gemm1(const float* __restrict__ X, const float* __restrict__ W1,
                        const float* __restrict__ dinv,
                        float* __restrict__ hs, float* __restrict__ acc1) {
    const int lane  = threadIdx.x & 31;
    const int wave  = threadIdx.x >> 5;
    const int tile  = blockIdx.x * 8 + wave;
    const int nTile = (N_NODES / 16) * (D_HID / 16);  // 12500
    if (tile >= nTile) return;                         // whole wave exits; EXEC stays full
    const int rt    = tile / (D_HID / 16);
    const int ct    = tile % (D_HID / 16);
    const int lrow  = lane & 15;
    const int khalf = lane >> 4;                       // 0: K={0,1}, 1: K={2,3}
    const int row   = rt * 16 + lrow;
    const int col   = ct * 16 + lrow;

    v8f acc = {};
    const float* xrow = X + (size_t)row * D_IN;
#pragma unroll
    for (int k0 = 0; k0 < D_IN; k0 += 4) {
        const int ka = k0 + khalf * 2;
        v2f a; a.x = xrow[ka];                a.y = xrow[ka + 1];
        v2f b; b.x = W1[ka * D_HID + col];    b.y = W1[(ka + 1) * D_HID + col];
        acc = __builtin_amdgcn_wmma_f32_16x16x4_f32(false, a, false, b,
                                                    (short)0, acc, false, false);
    }
#pragma unroll
    for (int i = 0; i < 8; ++i) {
        const int r  = rt * 16 + khalf * 8 + i;        // C/D layout: M = i + 8*khalf
        const float dv = dinv[r];
        const float v  = acc[i];
        hs  [(size_t)r * D_HID + col] = v * dv;
        acc1[(size_t)r * D_HID + col] = v * dv * dv;
    }
}

// ------------------------------------------------ scatter1: acc1[d] += ... --
// One edge per wave32, 2 feats per lane (D_HID=64). hs already holds h0*dinv[s].
__global__ void k_scatter1(const long long* __restrict__ src, const long long* __restrict__ dst,
                           const float* __restrict__ dinv, const float* __restrict__ hs,
                           float* __restrict__ acc1, int E) {
    const int gid  = blockIdx.x * blockDim.x + threadIdx.x;
    const int e    = gid >> 5;
    if (e >= E) return;
    const int lane = gid & 31;
    __builtin_prefetch(src + e + 4096, 0, 1);
    const int s = (int)src[e];
    const int d = (int)dst[e];
    const float dv = dinv[d];
    const float2 v = ((const float2*)(hs + (size_t)s * D_HID))[lane];
    float* ap = acc1 + (size_t)d * D_HID + lane * 2;
    atomicAdd(ap,     v.x * dv);
    atomicAdd(ap + 1, v.y * dv);
}

// --------------------- GEMM2 (fused): x1 = h@Wl+bl, h2 = relu(h)@W2 ---------
// Writes d_out = x + x1 + b2 + h2*dinv^2 (self-loop seed) and hs2 = h2*dinv.
__global__ void k_gemm2(const float* __restrict__ acc1, const float* __restrict__ b1,
                        const float* __restrict__ Wl,   const float* __restrict__ bl,
                        const float* __restrict__ W2,   const float* __restrict__ b2,
                        const float* __restrict__ X,    const float* __restrict__ dinv,
                        float* __restrict__ out, float* __restrict__ hs2) {
    const int lane  = threadIdx.x & 31;
    const int wave  = threadIdx.x >> 5;
    const int tile  = blockIdx.x * 8 + wave;
    const int nTile = (N_NODES / 16) * (D_OUT / 16);   // 25000
    if (tile >= nTile) return;
    const int rt    = tile / (D_OUT / 16);
    const int ct    = tile % (D_OUT / 16);
    const int lrow  = lane & 15;
    const int khalf = lane >> 4;
    const int row   = rt * 16 + lrow;
    const int col   = ct * 16 + lrow;
    const float blc = bl[col];
    const float b2c = b2[col];

    v8f accL = {};   // h @ Wl
    v8f accR = {};   // relu(h) @ W2
    const float* hrow = acc1 + (size_t)row * D_HID;
#pragma unroll
    for (int k0 = 0; k0 < D_HID; k0 += 4) {
        const int ka = k0 + khalf * 2;
        const float h0 = hrow[ka]     + b1[ka];
        const float h1 = hrow[ka + 1] + b1[ka + 1];
        v2f aL; aL.x = h0;              aL.y = h1;
        v2f aR; aR.x = fmaxf(h0, 0.f);  aR.y = fmaxf(h1, 0.f);
        v2f bL; bL.x = Wl[ka * D_OUT + col]; bL.y = Wl[(ka + 1) * D_OUT + col];
        v2f bR; bR.x = W2[ka * D_OUT + col]; bR.y = W2[(ka + 1) * D_OUT + col];
        accL = __builtin_amdgcn_wmma_f32_16x16x4_f32(false, aL, false, bL,
                                                     (short)0, accL, false, false);
        accR = __builtin_amdgcn_wmma_f32_16x16x4_f32(false, aR, false, bR,
                                                     (short)0, accR, false, false);
    }
#pragma unroll
    for (int i = 0; i < 8; ++i) {
        const int r  = rt * 16 + khalf * 8 + i;
        const float dv = dinv[r];
        const float x1 = accL[i] + blc;
        const float h2 = accR[i];
        out[(size_t)r * D_OUT + col] = X[(size_t)r * D_IN + col] + x1 + b2c + h2 * dv * dv;
        hs2[(size_t)r * D_OUT + col] = h2 * dv;
    }
}

// ------------------------------------------------ scatter2: out[d] += ... ---
// One edge per wave32, 4 feats per lane (D_OUT=128). hs2 holds h2*dinv[s].
__global__ void k_scatter2(const long long* __restrict__ src, const long long* __restrict__ dst,
                           const float* __restrict__ dinv, const float* __restrict__ hs2,
                           float* __restrict__ out, int E) {
    const int gid  = blockIdx.x * blockDim.x + threadIdx.x;
    const int e    = gid >> 5;
    if (e >= E) return;
    const int lane = gid & 31;
    __builtin_prefetch(src + e + 4096, 0, 1);
    const int s = (int)src[e];
    const int d = (int)dst[e];
    const float dv = dinv[d];
    const float4 v = ((const float4*)(hs2 + (size_t)s * D_OUT))[lane];
    float* ap = out + (size_t)d * D_OUT + lane * 4;
    atomicAdd(ap,     v.x * dv);
    atomicAdd(ap + 1, v.y * dv);
    atomicAdd(ap + 2, v.z * dv);
    atomicAdd(ap + 3, v.w * dv);
}

// ---------------------------------------------------------------------------
extern "C" void kernel_launch(void* const* d_in, const int* in_sizes, int n_in,
                              void* d_out, int out_size, void* d_ws, size_t ws_size,
                              hipStream_t stream) {
    const float*     x   = (const float*)d_in[0];
    const long long* ei  = (const long long*)d_in[1];
    const float*     W1  = (const float*)d_in[2];
    const float*     b1  = (const float*)d_in[3];
    const float*     Wl  = (const float*)d_in[4];
    const float*     bl  = (const float*)d_in[5];
    const float*     W2  = (const float*)d_in[6];
    const float*     b2  = (const float*)d_in[7];
    float* out = (float*)d_out;

    const int N = N_NODES;
    const int E = in_sizes[1] / 2;          // edge_index is [2, E] int64
    const long long* src = ei;
    const long long* dst = ei + E;

    // scratch layout (fp32): deg[N] | dinv[N] | hs[N*64] | acc1[N*64] | hs2[N*128]
    float* ws   = (float*)d_ws;
    float* deg  = ws;
    float* dinv = deg  + N;
    float* hs   = dinv + N;
    float* acc1 = hs   + (size_t)N * D_HID;
    float* hs2  = acc1 + (size_t)N * D_HID;

    k_init_deg<<<(N + 255) / 256, 256, 0, stream>>>(deg, N);
    k_deg     <<<(E + 255) / 256, 256, 0, stream>>>(dst, deg, E);
    k_dinv    <<<(N + 255) / 256, 256, 0, stream>>>(deg, dinv, N);

    const int t1 = (N / 16) * (D_HID / 16);   // 12500 wave-tiles
    k_gemm1<<<(t1 + 7) / 8, 256, 0, stream>>>(x, W1, dinv, hs, acc1);

    k_scatter1<<<(E * 32 + 255) / 256, 256, 0, stream>>>(src, dst, dinv, hs, acc1, E);

    const int t2 = (N / 16) * (D_OUT / 16);   // 25000 wave-tiles
    k_gemm2<<<(t2 + 7) / 8, 256, 0, stream>>>(acc1, b1, Wl, bl, W2, b2, x, dinv, out, hs2);

    k_scatter2<<<(E * 32 + 255) / 256, 256, 0, stream>>>(src, dst, dinv, hs2, out, E);
}